// Resonance_22016002360005
// MI455X (gfx1250) — compile-verified
//
#include <hip/hip_runtime.h>
#include <stdint.h>

// Problem constants from the reference.
#define BATCH      4
#define N_EVENTS   64
#define N_RES      4096
#define N_SAMPLES  32768
#define N_BE       (BATCH * N_EVENTS)   // 256 (b,e) pairs

// TDM copy tiling: 128KB row copied in 4 chunks of 32KB, double-buffered in LDS.
#define CHUNK_ELTS 8192                  // floats per chunk (32KB)
#define N_CHUNKS   (N_SAMPLES / CHUNK_ELTS)

typedef __attribute__((ext_vector_type(4))) unsigned int v4u;
typedef __attribute__((ext_vector_type(8))) int          v8i;
typedef __attribute__((ext_vector_type(4))) int          v4i;

// ---------------------------------------------------------------------------
// TDM descriptor builders (CDNA5 ISA ch. 8, D# groups 0/1).
// Group 0: [1:0]=count=1, [63:32]=lds_addr, [120:64]=global_addr, [127:126]=2.
// Group 1: data_size=2 (4B); tensor_dim0=tile_dim0=n; tensor_dim1=tile_dim1=1;
//          tensor_dim0_stride=n; no padding/iterate/multicast/barrier.
// ---------------------------------------------------------------------------
__device__ __forceinline__ v4u tdm_g0(unsigned lds_off, unsigned long long gaddr) {
    v4u g;
    g.x = 1u;                                       // count=1 (valid user D#)
    g.y = lds_off;                                  // LDS byte address
    g.z = (unsigned)gaddr;                          // global addr [95:64]
    g.w = (unsigned)(gaddr >> 32) | 0x80000000u;    // addr [120:96] | type=2
    return g;
}

__device__ __forceinline__ v8i tdm_g1(unsigned n_elts) {
    v8i g;
    g[0] = 0x00020000;                  // data_size = 2 -> 4-byte elements
    g[1] = (int)(n_elts << 16);         // tensor_dim0[15:0] at bits [63:48]
    g[2] = 0x00010000 |                 // tensor_dim1 = 1  at bits [95:80]
           (int)(n_elts >> 16);         // tensor_dim0[31:16] at bits [79:64]
    g[3] = (int)(n_elts << 16);         // tile_dim0 at bits [127:112]
    g[4] = 1;                           // tile_dim1 = 1, tile_dim2 = 0
    g[5] = (int)n_elts;                 // tensor_dim0_stride[31:0]
    g[6] = 0;                           // stride[47:32]=0, dim1_stride lo = 0
    g[7] = 0;
    return g;
}

// ---------------------------------------------------------------------------
// Kernel A (emitted first so the disasm snippet shows it): one wave per
// (b,e). DMA the selected 128KB waveform row to the output row via the
// Tensor Data Mover: global -> LDS -> global, 32KB chunks, double-buffered,
// pipelining store(k) with load(k+1) under TENSORcnt.
// Builtin signature on this toolchain (6 args):
//   (v4u g0, v8i g1, v4i, v4i, v8i, i32 cpol)
// Trailing groups are zero (2D tensor; unused).
// ---------------------------------------------------------------------------
__global__ __launch_bounds__(32) void
tdm_gather_kernel(const float* __restrict__ waves,
                  const int*   __restrict__ idx_in,
                  float*       __restrict__ out) {
    __shared__ float lbuf[2][CHUNK_ELTS];   // 2 x 32KB staging buffers

    const int be  = blockIdx.x;
    int idx = idx_in[be];
    idx = __builtin_amdgcn_readfirstlane(idx);

    const unsigned long long src =
        (unsigned long long)(uintptr_t)(waves + (size_t)idx * N_SAMPLES);
    const unsigned long long dst =
        (unsigned long long)(uintptr_t)(out + (size_t)be * N_SAMPLES);

    const unsigned lds0 = (unsigned)(uintptr_t)&lbuf[0][0];
    const unsigned lds1 = (unsigned)(uintptr_t)&lbuf[1][0];

    const v8i g1  = tdm_g1(CHUNK_ELTS);
    const v4i gz4 = (v4i){0, 0, 0, 0};      // groups 2/3 unused (2D tensor)
    const v8i gz8 = (v8i){0, 0, 0, 0, 0, 0, 0, 0};

    // Prologue: load chunk 0 into buffer 0.
    __builtin_amdgcn_tensor_load_to_lds(tdm_g0(lds0, src), g1, gz4, gz4, gz8, 0);

    for (int k = 0; k < N_CHUNKS; ++k) {
        // Tensor ops from one wave complete in order: waiting TENSORcnt<=0
        // joins both the previous store and the load of chunk k.
        __builtin_amdgcn_s_wait_tensorcnt(0);

        const unsigned lds_cur = (k & 1) ? lds1 : lds0;
        __builtin_amdgcn_tensor_store_from_lds(
            tdm_g0(lds_cur, dst + (unsigned long long)k * CHUNK_ELTS * 4),
            g1, gz4, gz4, gz8, 0);

        if (k + 1 < N_CHUNKS) {
            const unsigned lds_nxt = ((k + 1) & 1) ? lds1 : lds0;
            __builtin_amdgcn_tensor_load_to_lds(
                tdm_g0(lds_nxt, src + (unsigned long long)(k + 1) * CHUNK_ELTS * 4),
                g1, gz4, gz4, gz8, 0);
        }
    }
    __builtin_amdgcn_s_wait_tensorcnt(0);   // drain final store
}

// ---------------------------------------------------------------------------
// Kernel B: argmax over the resonance axis. 256 blocks (one per (b,e)),
// 256 threads each scan 16 strided elements, then LDS tree reduce.
// Tie-break: lowest index (matches jnp.argmax).
// ---------------------------------------------------------------------------
__global__ __launch_bounds__(256) void
argmax_kernel(const float* __restrict__ choice, int* __restrict__ idx_out) {
    const int be = blockIdx.x;
    const float* row = choice + (size_t)be * N_RES;
    const int t = threadIdx.x;

    float best = -__builtin_inff();
    int   bidx = N_RES;  // larger than any valid index

    for (int r = t; r < N_RES; r += 256) {
        float v = row[r];
        if (v > best) { best = v; bidx = r; }
        // strided ascending scan => first hit of a value is the lowest index
    }

    __shared__ float sval[256];
    __shared__ int   sidx[256];
    sval[t] = best;
    sidx[t] = bidx;
    __syncthreads();

    for (int s = 128; s > 0; s >>= 1) {
        if (t < s) {
            float v2 = sval[t + s];
            int   i2 = sidx[t + s];
            if (v2 > sval[t] || (v2 == sval[t] && i2 < sidx[t])) {
                sval[t] = v2;
                sidx[t] = i2;
            }
        }
        __syncthreads();
    }
    if (t == 0) idx_out[be] = sidx[0];
}

// ---------------------------------------------------------------------------
// Launch: argmax (choice -> d_ws indices), then TDM row gather.
// ---------------------------------------------------------------------------
extern "C" void kernel_launch(void* const* d_in, const int* in_sizes, int n_in,
                              void* d_out, int out_size, void* d_ws, size_t ws_size,
                              hipStream_t stream) {
    const float* choice = (const float*)d_in[0];   // [4, 64, 4096] f32
    const float* waves  = (const float*)d_in[1];   // [4096, 32768] f32
    float*       out    = (float*)d_out;           // [4, 64, 32768] f32
    int*         idxbuf = (int*)d_ws;              // 256 ints of scratch

    argmax_kernel<<<N_BE, 256, 0, stream>>>(choice, idxbuf);
    tdm_gather_kernel<<<N_BE, 32, 0, stream>>>(waves, idxbuf, out);
}